// ESNModel_35399120453975
// MI455X (gfx1250) — compile-verified
//
#include <hip/hip_runtime.h>
#include <math.h>

// ---------------------------------------------------------------------------
// ESN (leaky echo-state network), B=8 S=128 N=64 F=32 H=256 L=3, fp32.
// fp32 WMMA (V_WMMA_F32_16X16X4_F32) everywhere; W_h DMA'd into CDNA5's
// 320KB LDS via the Tensor Data Mover (with HW row padding for bank-conflict-
// free fragment reads); 512 independent recurrences -> 32 blocks x 16 rows.
// ---------------------------------------------------------------------------

typedef __attribute__((ext_vector_type(2))) float v2f;
typedef __attribute__((ext_vector_type(8))) float v8f;
typedef __attribute__((ext_vector_type(4))) unsigned int u32x4;
typedef __attribute__((ext_vector_type(4))) int i32x4;
typedef __attribute__((ext_vector_type(8))) int i32x8;

#define LEAK   0.9f
#define S_     128
#define N_     64
#define H_     256
#define M_     65536          // B*S*N rows
#define OUTF   768            // L*H output feature stride
#define WH_STRIDE 260         // LDS row pad: 260 % 64 == 4 -> conflict-free b64

__device__ __forceinline__ v2f ld2(const float* p) { return *(const v2f*)p; }

// ---------------------------------------------------------------------------
// U = X @ W^T + bias.  One wave computes a 16x64 tile (4 WMMA column tiles).
// grid = 2048 blocks * 8 waves = 16384 waves = 4096 m-tiles * 4 col-quads.
// ---------------------------------------------------------------------------
__global__ __launch_bounds__(256) void esn_input_gemm(
    const float* __restrict__ X, int xstride, int K,
    const float* __restrict__ W,      // [256, K] row-major (output-major)
    const float* __restrict__ bias,   // [256]
    float* __restrict__ U)            // [M_, 256]
{
  const int tid  = threadIdx.x;
  const int lane = tid & 31, wave = tid >> 5;
  const int half = lane >> 4, l15 = lane & 15;
  const int wg     = blockIdx.x * 8 + wave;
  const int m_tile = wg >> 2;
  const int nq     = wg & 3;
  const int m0     = m_tile * 16;
  const int nb     = nq * 64;
  const int koff   = 2 * half;          // A/B fragment K split across lane halves

  const float* Xrow  = X + (size_t)(m0 + l15) * xstride;
  const float* Wrow0 = W + (size_t)(nb +  0 + l15) * K;
  const float* Wrow1 = W + (size_t)(nb + 16 + l15) * K;
  const float* Wrow2 = W + (size_t)(nb + 32 + l15) * K;
  const float* Wrow3 = W + (size_t)(nb + 48 + l15) * K;

  // init accumulators with bias (bias is constant along M within a tile)
  float bv0 = bias[nb +  0 + l15];
  float bv1 = bias[nb + 16 + l15];
  float bv2 = bias[nb + 32 + l15];
  float bv3 = bias[nb + 48 + l15];
  v8f c0, c1, c2, c3;
#pragma unroll
  for (int j = 0; j < 8; ++j) { c0[j] = bv0; c1[j] = bv1; c2[j] = bv2; c3[j] = bv3; }

#pragma unroll 4
  for (int k0 = 0; k0 < K; k0 += 4) {
    v2f a  = ld2(Xrow  + k0 + koff);
    v2f w0 = ld2(Wrow0 + k0 + koff);
    v2f w1 = ld2(Wrow1 + k0 + koff);
    v2f w2 = ld2(Wrow2 + k0 + koff);
    v2f w3 = ld2(Wrow3 + k0 + koff);
    c0 = __builtin_amdgcn_wmma_f32_16x16x4_f32(false, a, false, w0, (short)0, c0, false, false);
    c1 = __builtin_amdgcn_wmma_f32_16x16x4_f32(false, a, false, w1, (short)0, c1, false, false);
    c2 = __builtin_amdgcn_wmma_f32_16x16x4_f32(false, a, false, w2, (short)0, c2, false, false);
    c3 = __builtin_amdgcn_wmma_f32_16x16x4_f32(false, a, false, w3, (short)0, c3, false, false);
  }

#pragma unroll
  for (int j = 0; j < 8; ++j) {
    int row = m0 + j + 8 * half;        // C/D layout: rows j / j+8 by lane half
    float* Urow = U + (size_t)row * 256 + l15;
    Urow[nb +  0] = c0[j];
    Urow[nb + 16] = c1[j];
    Urow[nb + 32] = c2[j];
    Urow[nb + 48] = c3[j];
  }
}

// ---------------------------------------------------------------------------
// Scan: h <- (1-a)h + a*tanh(u_t + h @ Wh^T).  grid = 32 blocks; block owns
// 16 (b,n) rows.  Wh lives in LDS (256x260 f32, 266KB, TDM-loaded with HW
// padding), H tile in LDS.  8 waves x two 16x16 column tiles per step.
// ---------------------------------------------------------------------------
__global__ __launch_bounds__(256) void esn_scan(
    const float* __restrict__ U,      // [M_, 256]
    const float* __restrict__ Wh,     // [256, 256] row-major [g][h]
    float* __restrict__ out,          // [M_, OUTF]
    int layer)
{
  extern __shared__ float lds[];
  float* Hs = lds;                       // 16  x WH_STRIDE
  float* Ws = lds + 16 * WH_STRIDE;      // 256 x WH_STRIDE

  const int tid = threadIdx.x;

#if defined(__has_builtin) && __has_builtin(__builtin_amdgcn_tensor_load_to_lds)
  // ---- Tensor Data Mover: DMA Wh (256x256 f32) -> LDS, padding each 256-
  // DWORD row by 4 DWORDs (pad_interval code 7 = 256, pad_amount code 3 = 4)
  // to produce the 260-float conflict-free row stride in hardware.
  if (tid < 32) {                        // one wave issues the DMA
    unsigned long long ga = (unsigned long long)(uintptr_t)Wh;
    unsigned int lds_addr = (unsigned int)(uintptr_t)(void*)Ws;  // low 32 bits = LDS byte offset
    u32x4 g0 = { 1u,                                   // count=1 (valid D#)
                 lds_addr,                             // lds_addr [63:32]
                 (unsigned int)(ga & 0xFFFFFFFFu),     // global_addr [95:64]
                 (unsigned int)((ga >> 32) & 0x01FFFFFFu) | (2u << 30) }; // [120:96] | type=2
    i32x8 g1 = { (int)((2u << 16) | (1u << 20) | (7u << 22) | (3u << 25)),
                 //  data_size=4B    pad_en      interval=256  amount=4
                 (int)(256u << 16),    // tensor_dim0[15:0] at bits 63:48
                 (int)(256u << 16),    // tensor_dim1[15:0] at bits 111:80 (lo16 here)
                 (int)(256u << 16),    // tile_dim0 at bits 127:112
                 (int)256,             // tile_dim1 at bits 143:128
                 (int)256,             // tensor_dim0_stride[31:0] at bits 207:160
                 0, 0 };
    i32x4 gz = { 0, 0, 0, 0 };
#if __clang_major__ >= 23
    i32x8 gz8 = { 0, 0, 0, 0, 0, 0, 0, 0 };
    __builtin_amdgcn_tensor_load_to_lds(g0, g1, gz, gz, gz8, 0);
#else
    __builtin_amdgcn_tensor_load_to_lds(g0, g1, gz, gz, 0);
#endif
    __builtin_amdgcn_s_wait_tensorcnt(0);
  }
#else
  for (int i = tid; i < 256 * 256; i += 256) {
    int r = i >> 8, cc = i & 255;
    Ws[r * WH_STRIDE + cc] = Wh[i];      // B[k][g] = Wh[g][k] -> rows indexed by g
  }
#endif
  for (int i = tid; i < 16 * WH_STRIDE; i += 256) Hs[i] = 0.0f;
  __syncthreads();

  const int lane = tid & 31, wave = tid >> 5;
  const int half = lane >> 4, l15 = lane & 15;
  const int col0 = wave * 32 + l15;
  const int col1 = col0 + 16;
  const int p_base = blockIdx.x * 16;
  const int koff = 2 * half;

  int rowoff[8];
#pragma unroll
  for (int j = 0; j < 8; ++j) {
    int p = p_base + j + 8 * half;       // flattened (b,n) pair this C element maps to
    int b = p >> 6, n = p & 63;
    rowoff[j] = b * (S_ * N_) + n;       // global row at time t = rowoff + t*N_
  }

  const float* HsA  = Hs + l15  * WH_STRIDE + koff;   // A frag: M=l15, K pair by half
  const float* WsB0 = Ws + col0 * WH_STRIDE + koff;   // B frag: N=col, K pair by half
  const float* WsB1 = Ws + col1 * WH_STRIDE + koff;

  for (int t = 0; t < S_; ++t) {
    v8f c0, c1;
#pragma unroll
    for (int j = 0; j < 8; ++j) {        // seed accumulators with u_t
      const float* urow = U + (size_t)(rowoff[j] + t * N_) * 256;
      c0[j] = urow[col0];
      c1[j] = urow[col1];
      if (t + 1 < S_) {                  // hide next step's u_t latency
        __builtin_prefetch(urow + (size_t)N_ * 256 + col0, 0, 0);
      }
    }
#pragma unroll 4
    for (int k0 = 0; k0 < H_; k0 += 4) {
      v2f a  = ld2(HsA  + k0);
      v2f b0 = ld2(WsB0 + k0);
      v2f b1 = ld2(WsB1 + k0);
      c0 = __builtin_amdgcn_wmma_f32_16x16x4_f32(false, a, false, b0, (short)0, c0, false, false);
      c1 = __builtin_amdgcn_wmma_f32_16x16x4_f32(false, a, false, b1, (short)0, c1, false, false);
    }
    float hn0[8], hn1[8];
#pragma unroll
    for (int j = 0; j < 8; ++j) {
      int m = j + 8 * half;
      hn0[j] = (1.0f - LEAK) * Hs[m * WH_STRIDE + col0] + LEAK * tanhf(c0[j]);
      hn1[j] = (1.0f - LEAK) * Hs[m * WH_STRIDE + col1] + LEAK * tanhf(c1[j]);
    }
    __syncthreads();                     // all reads of Hs done
#pragma unroll
    for (int j = 0; j < 8; ++j) {
      int m = j + 8 * half;
      Hs[m * WH_STRIDE + col0] = hn0[j];
      Hs[m * WH_STRIDE + col1] = hn1[j];
      float* orow = out + (size_t)(rowoff[j] + t * N_) * OUTF + layer * H_;
      orow[col0] = hn0[j];
      orow[col1] = hn1[j];
    }
    __syncthreads();                     // Hs update visible before next step
  }
}

// ---------------------------------------------------------------------------
// host launcher
// ---------------------------------------------------------------------------
extern "C" void kernel_launch(void* const* d_in, const int* in_sizes, int n_in,
                              void* d_out, int out_size, void* d_ws, size_t ws_size,
                              hipStream_t stream) {
  const float* x     = (const float*)d_in[0];  // [8,128,64,32]
  const float* w_ih0 = (const float*)d_in[1];  // [256,32]
  const float* w_ih  = (const float*)d_in[2];  // [2,256,256]
  const float* b_ih  = (const float*)d_in[3];  // [3,256]
  const float* w_hh  = (const float*)d_in[4];  // [3,256,256]
  // d_in[5] = time_scale == 1 (output reshape is a no-op on flat memory)
  float* out = (float*)d_out;                  // [65536, 768]
  float* U   = (float*)d_ws;                   // [65536, 256] = 64 MB scratch

  const size_t shmem = (size_t)(16 * WH_STRIDE + 256 * WH_STRIDE) * sizeof(float);
  (void)hipFuncSetAttribute(reinterpret_cast<const void*>(esn_scan),
                            hipFuncAttributeMaxDynamicSharedMemorySize, (int)shmem);

  for (int l = 0; l < 3; ++l) {
    const float* X; int xs, K; const float* W;
    if (l == 0) { X = x;                xs = 32;   K = 32; W = w_ih0; }
    else        { X = out + (l-1) * H_; xs = OUTF; K = H_; W = w_ih + (size_t)(l-1) * H_ * H_; }
    esn_input_gemm<<<2048, 256, 0, stream>>>(X, xs, K, W, b_ih + l * H_, U);
    esn_scan<<<32, 256, shmem, stream>>>(U, w_hh + (size_t)l * H_ * H_, out, l);
  }
}